// Model3DECGAT_19181323944000
// MI455X (gfx1250) — compile-verified
//
#include <hip/hip_runtime.h>
#include <hip/hip_bf16.h>
#include <math.h>

// ---------------- problem constants (from reference setup_inputs) ----------
#define B_    2
#define W_    16
#define N_    1000
#define FIN   16
#define E_    16000
#define DE    8
#define H_    128
#define HEADS 4
#define DK    32
#define L_    3
#define NT    (W_ * N_)                 // 16000 nodes per batch
#define ET    (W_ * E_ + (W_ - 1) * N_) // 271000 edges
#define ROWS  (B_ * NT)                 // 32000 node rows (batches contiguous)

typedef __attribute__((ext_vector_type(2))) float v2f;
typedef __attribute__((ext_vector_type(8))) float v8f;
typedef int v4i_gcc __attribute__((vector_size(16)));   // matches builtin proto
typedef __attribute__((address_space(1))) v4i_gcc* gptr_v4i;
typedef __attribute__((address_space(3))) v4i_gcc* lptr_v4i;

// ---------------------------------------------------------------------------
// async staging helpers (gfx1250 GLOBAL_LOAD_ASYNC_TO_LDS_* path, ASYNCcnt)
// ---------------------------------------------------------------------------
__device__ __forceinline__ void stage_b128(const float* gsrc, float* lds_dst) {
#if __has_builtin(__builtin_amdgcn_global_load_async_to_lds_b128)
    float* g = const_cast<float*>(gsrc);               // drop const first
    __builtin_amdgcn_global_load_async_to_lds_b128(
        (gptr_v4i)g, (lptr_v4i)lds_dst, /*offset=*/0, /*cpol=*/0);
#else
    *(float4*)lds_dst = *(const float4*)gsrc;          // synchronous fallback
#endif
}

__device__ __forceinline__ void wait_async_copies() {
#if __has_builtin(__builtin_amdgcn_global_load_async_to_lds_b128)
#if __has_builtin(__builtin_amdgcn_s_wait_asynccnt)
    __builtin_amdgcn_s_wait_asynccnt(0);
#else
    asm volatile("s_wait_asynccnt 0x0" ::: "memory");
#endif
#endif
}

// ---------------------------------------------------------------------------
// fp32 WMMA GEMM: C[M,N] = A[M,K] @ W[K,N] + bias[N]
// block = 128 threads (4 waves) -> 16 rows x 128 cols of C. Requires
// M % 16 == 0, N % 128 == 0, K % 4 == 0, K <= 128 (true for all our shapes).
//
// The 16xK A-panel (contiguous in memory: full-width rows) is staged to LDS
// once per block with GLOBAL_LOAD_ASYNC_TO_LDS_B128 + s_wait_asynccnt, padded
// to a (K+4)-float row pitch so the per-k A-frag ds_load_b64 is conflict-free
// (lane l low half -> banks 4l..4l+1, high half -> 4l+2..4l+3; all 64 banks).
// Each wave owns a 16x32 C tile: 2 WMMA accumulators sharing one A-frag.
//
// Fragment layouts (ISA 7.12.2, V_WMMA_F32_16X16X4_F32):
//  A (16x4): lanes 0-15: M=lane, K={k,k+1}; lanes 16-31: M=lane-16, K={k+2,k+3}
//  B (4x16): VGPR j: lanes 0-15 -> K=k+j, N=lane; lanes 16-31 -> K=k+j+2
//  C/D:      VGPR j: lanes 0-15 -> M=j, N=lane;   lanes 16-31 -> M=j+8
// ---------------------------------------------------------------------------
__global__ void gemm_wmma_f32(const float* __restrict__ A,
                              const float* __restrict__ Wt,
                              const float* __restrict__ bias,
                              float* __restrict__ C,
                              int M, int K, int N) {
    __shared__ float tA[16 * (128 + 4)];
    const int KP = K + 4;                       // padded LDS row pitch (floats)
    int tid  = threadIdx.x;
    int lane = tid & 31;
    int wave = tid >> 5;
    int m0 = blockIdx.y * 16;
    int n0 = blockIdx.x * 128 + wave * 32;      // wave's 32-col slice

    // ---- stage A panel (rows m0..m0+15, cols 0..K) into LDS ----
    const int K4 = K >> 2;                      // float4 chunks per row
    for (int i = tid; i < 16 * K4; i += 128) {
        int r = i / K4;
        int c = (i - r * K4) << 2;
        stage_b128(A + (size_t)(m0 + r) * K + c, tA + r * KP + c);
    }
    wait_async_copies();
    __syncthreads();

    int hi = lane >> 4;                         // 0 = lanes 0-15, 1 = 16-31
    int lm = lane & 15;

    v8f acc0 = {0.f, 0.f, 0.f, 0.f, 0.f, 0.f, 0.f, 0.f};
    v8f acc1 = {0.f, 0.f, 0.f, 0.f, 0.f, 0.f, 0.f, 0.f};
    for (int k = 0; k < K; k += 4) {
        int ka = k + hi * 2;
        v2f a;
        a.x = tA[lm * KP + ka];                 // ds_load_b64, conflict-free
        a.y = tA[lm * KP + ka + 1];
        const float* bp = Wt + (size_t)ka * N + n0 + lm;
        v2f b0, b1;
        b0.x = bp[0];       b0.y = bp[N];
        b1.x = bp[16];      b1.y = bp[N + 16];
        acc0 = __builtin_amdgcn_wmma_f32_16x16x4_f32(
            false, a, false, b0, (short)0, acc0, false, false);
        acc1 = __builtin_amdgcn_wmma_f32_16x16x4_f32(
            false, a, false, b1, (short)0, acc1, false, false);
    }

    float bv0 = bias[n0 + lm];
    float bv1 = bias[n0 + 16 + lm];
#pragma unroll
    for (int j = 0; j < 8; ++j) {
        int row = m0 + j + hi * 8;
        C[(size_t)row * N + n0 + lm]      = acc0[j] + bv0;
        C[(size_t)row * N + n0 + 16 + lm] = acc1[j] + bv1;
    }
}

// ---------------------------------------------------------------------------
// helpers
// ---------------------------------------------------------------------------
__device__ __forceinline__ void atomicMaxFloat(float* addr, float val) {
    // valid with init = -inf; monotone int trick for mixed signs
    if (val >= 0.f) atomicMax((int*)addr, __float_as_int(val));
    else            atomicMin((unsigned int*)addr, (unsigned int)__float_as_int(val));
}

__device__ __forceinline__ void decode_edge(int e, const int* __restrict__ sp_ei,
                                            int& src, int& dst, int& sp_j) {
    if (e < W_ * E_) {
        int w = e / E_;
        int j = e - w * E_;
        src = sp_ei[j] + w * N_;
        dst = sp_ei[E_ + j] + w * N_;
        sp_j = j;
    } else {
        int i = e - W_ * E_;
        src = i;
        dst = i + N_;
        sp_j = -1;          // temporal edge -> use mean edge attr
    }
}

// mean of spatial_ea over E rows -> ea_mean[8]
__global__ void ea_mean_kernel(const float* __restrict__ ea, float* __restrict__ out) {
    __shared__ float red[256];
    int c = blockIdx.x;                       // column 0..7
    float s = 0.f;
    for (int i = threadIdx.x; i < E_; i += 256) s += ea[(size_t)i * DE + c];
    red[threadIdx.x] = s;
    __syncthreads();
    for (int off = 128; off; off >>= 1) {
        if ((int)threadIdx.x < off) red[threadIdx.x] += red[threadIdx.x + off];
        __syncthreads();
    }
    if (threadIdx.x == 0) out[c] = red[0] / (float)E_;
}

__global__ void init_ms_kernel(float* __restrict__ m, float* __restrict__ s, int n) {
    int i = blockIdx.x * 256 + threadIdx.x;
    if (i < n) { m[i] = -__builtin_inff(); s[i] = 0.f; }
}

__global__ void zero_kernel(float* __restrict__ p, int n) {
    int i = blockIdx.x * 256 + threadIdx.x;
    if (i < n) p[i] = 0.f;
}

// ---------------------------------------------------------------------------
// Pass A: one wave per (b, e, h); lane = dk index.
// k_e = K[src]*(1+gamma)+beta with FiLM recomputed on the fly (fw tiny, L2-hot)
// logit = <q_e, k_e>/sqrt(dk); store logit, atomic-max into m[b,dst,h].
// ---------------------------------------------------------------------------
__global__ void edge_logits_kernel(const float* __restrict__ Q, const float* __restrict__ K,
                                   const int* __restrict__ sp_ei, const float* __restrict__ sp_ea,
                                   const float* __restrict__ ea_mean,
                                   const float* __restrict__ fw, const float* __restrict__ fb,
                                   float* __restrict__ lg, float* __restrict__ mtab) {
    int wid = blockIdx.x * 4 + (threadIdx.x >> 5);
    int lane = threadIdx.x & 31;
    if (wid >= B_ * ET * HEADS) return;
    int h = wid & 3;
    int e = (wid >> 2) % ET;
    int b = wid / (4 * ET);
    int src, dst, sp_j;
    decode_edge(e, sp_ei, src, dst, sp_j);
    const float* ea = (sp_j >= 0) ? (sp_ea + (size_t)sp_j * DE) : ea_mean;

    int hd = h * DK + lane;
    float g  = fb[hd];
    float bt = fb[H_ + hd];
#pragma unroll
    for (int r = 0; r < DE; ++r) {
        float a = ea[r];
        g  += a * fw[r * (2 * H_) + hd];
        bt += a * fw[r * (2 * H_) + H_ + hd];
    }
    float kv = K[((size_t)b * NT + src) * H_ + hd];
    float qv = Q[((size_t)b * NT + dst) * H_ + hd];
    float p = qv * (kv * (1.f + g) + bt);
    for (int off = 16; off; off >>= 1) p += __shfl_xor(p, off, 32);
    if (lane == 0) {
        float l = p * 0.17677669529663687f;      // 1/sqrt(32)
        lg[wid] = l;
        atomicMaxFloat(&mtab[((size_t)b * NT + dst) * HEADS + h], l);
    }
}

// Pass B: ex = exp(logit - m[dst]); accumulate denominator
__global__ void edge_exp_kernel(float* __restrict__ lg, const float* __restrict__ mtab,
                                float* __restrict__ stab, const int* __restrict__ sp_ei) {
    int idx = blockIdx.x * 256 + threadIdx.x;
    if (idx >= B_ * ET * HEADS) return;
    int h = idx & 3;
    int e = (idx >> 2) % ET;
    int b = idx / (4 * ET);
    int src, dst, sp_j;
    decode_edge(e, sp_ei, src, dst, sp_j);
    float ex = expf(lg[idx] - mtab[((size_t)b * NT + dst) * HEADS + h]);
    lg[idx] = ex;
    atomicAdd(&stab[((size_t)b * NT + dst) * HEADS + h], ex);
}

// Pass C: scatter att * V[src] into out[b,dst,:]
__global__ void edge_scatter_kernel(const float* __restrict__ V, const float* __restrict__ lg,
                                    const float* __restrict__ stab, const int* __restrict__ sp_ei,
                                    float* __restrict__ attn) {
    int wid = blockIdx.x * 4 + (threadIdx.x >> 5);
    int lane = threadIdx.x & 31;
    if (wid >= B_ * ET * HEADS) return;
    int h = wid & 3;
    int e = (wid >> 2) % ET;
    int b = wid / (4 * ET);
    int src, dst, sp_j;
    decode_edge(e, sp_ei, src, dst, sp_j);
    int hd = h * DK + lane;
    float att = lg[wid] / (stab[((size_t)b * NT + dst) * HEADS + h] + 1e-16f);
    float v = V[((size_t)b * NT + src) * H_ + hd];
    atomicAdd(&attn[((size_t)b * NT + dst) * H_ + hd], att * v);
}

// X = LayerNorm(X + Y) * g + b  (outer residual collapses since RES_SCALE==1)
__global__ void ln_residual_kernel(float* __restrict__ X, const float* __restrict__ Y,
                                   const float* __restrict__ g, const float* __restrict__ b) {
    int row = blockIdx.x * 4 + (threadIdx.x >> 5);
    int lane = threadIdx.x & 31;
    if (row >= ROWS) return;
    float* xr = X + (size_t)row * H_;
    const float* yr = Y + (size_t)row * H_;
    float v[4];
#pragma unroll
    for (int i = 0; i < 4; ++i) { int c = lane + 32 * i; v[i] = xr[c] + yr[c]; }
    float s = v[0] + v[1] + v[2] + v[3];
    for (int off = 16; off; off >>= 1) s += __shfl_xor(s, off, 32);
    float mu = s * (1.f / H_);
    float q = 0.f;
#pragma unroll
    for (int i = 0; i < 4; ++i) { float d = v[i] - mu; q += d * d; }
    for (int off = 16; off; off >>= 1) q += __shfl_xor(q, off, 32);
    float inv = rsqrtf(q * (1.f / H_) + 1e-5f);
#pragma unroll
    for (int i = 0; i < 4; ++i) {
        int c = lane + 32 * i;
        xr[c] = (v[i] - mu) * inv * g[c] + b[c];
    }
}

// head: softplus(last @ head_w + head_b); one wave per output element
__global__ void head_kernel(const float* __restrict__ X, const float* __restrict__ hw,
                            const float* __restrict__ hb, float* __restrict__ out) {
    int wid = blockIdx.x * 4 + (threadIdx.x >> 5);
    int lane = threadIdx.x & 31;
    if (wid >= B_ * N_) return;
    int b = wid / N_, n = wid % N_;
    const float* row = X + ((size_t)b * NT + (W_ - 1) * N_ + n) * H_;
    float p = 0.f;
#pragma unroll
    for (int i = 0; i < 4; ++i) { int c = lane + 32 * i; p += row[c] * hw[c]; }
    for (int off = 16; off; off >>= 1) p += __shfl_xor(p, off, 32);
    if (lane == 0) {
        float z = p + hb[0];
        out[wid] = log1pf(expf(-fabsf(z))) + fmaxf(z, 0.f);   // stable softplus
    }
}

// ---------------------------------------------------------------------------
extern "C" void kernel_launch(void* const* d_in, const int* in_sizes, int n_in,
                              void* d_out, int out_size, void* d_ws, size_t ws_size,
                              hipStream_t stream) {
    const float* xw     = (const float*)d_in[0];
    const int*   sp_ei  = (const int*)  d_in[1];
    const float* sp_ea  = (const float*)d_in[2];
    const float* in_w   = (const float*)d_in[3];
    const float* in_b   = (const float*)d_in[4];
    const float* qw     = (const float*)d_in[5];
    const float* qb     = (const float*)d_in[6];
    const float* kw     = (const float*)d_in[7];
    const float* kb     = (const float*)d_in[8];
    const float* vw     = (const float*)d_in[9];
    const float* vb     = (const float*)d_in[10];
    const float* fw     = (const float*)d_in[11];
    const float* fb     = (const float*)d_in[12];
    const float* ow     = (const float*)d_in[13];
    const float* ob     = (const float*)d_in[14];
    const float* ln_g   = (const float*)d_in[15];
    const float* ln_b   = (const float*)d_in[16];
    const float* head_w = (const float*)d_in[17];
    const float* head_b = (const float*)d_in[18];
    float* out = (float*)d_out;

    // workspace layout (floats)
    float* ws  = (float*)d_ws;
    const size_t SB = (size_t)ROWS * H_;          // 4,096,000
    float* X   = ws;                              // node features
    float* Qb_ = X  + SB;                         // Q (reused as Y after pass A)
    float* Kb_ = Qb_ + SB;
    float* Vb_ = Kb_ + SB;
    float* ATT = Vb_ + SB;                        // attention accumulator
    float* LG  = ATT + SB;                        // logits / exp, B*ET*HEADS
    float* Mx  = LG  + (size_t)B_ * ET * HEADS;
    float* Sx  = Mx  + (size_t)B_ * NT * HEADS;
    float* EAM = Sx  + (size_t)B_ * NT * HEADS;   // 8 floats

    const int nMS   = B_ * NT * HEADS;            // 128,000
    const int nEh   = B_ * ET * HEADS;            // 2,168,000 (multiple of 4)
    const int blkEw = nEh / 4;                    // wave-per-(b,e,h) blocks
    const dim3 gemmGrid(H_ / 128, ROWS / 16);     // (1, 2000)

    ea_mean_kernel<<<DE, 256, 0, stream>>>(sp_ea, EAM);

    // input projection: X = xw @ in_w + in_b   (K=16)
    gemm_wmma_f32<<<gemmGrid, 128, 0, stream>>>(xw, in_w, in_b, X, ROWS, FIN, H_);

    for (int l = 0; l < L_; ++l) {
        const float* qw_l = qw + (size_t)l * H_ * H_;
        const float* kw_l = kw + (size_t)l * H_ * H_;
        const float* vw_l = vw + (size_t)l * H_ * H_;
        const float* ow_l = ow + (size_t)l * H_ * H_;
        const float* fw_l = fw + (size_t)l * DE * 2 * H_;
        const float* fb_l = fb + (size_t)l * 2 * H_;

        gemm_wmma_f32<<<gemmGrid, 128, 0, stream>>>(X, qw_l, qb + l * H_, Qb_, ROWS, H_, H_);
        gemm_wmma_f32<<<gemmGrid, 128, 0, stream>>>(X, kw_l, kb + l * H_, Kb_, ROWS, H_, H_);
        gemm_wmma_f32<<<gemmGrid, 128, 0, stream>>>(X, vw_l, vb + l * H_, Vb_, ROWS, H_, H_);

        init_ms_kernel<<<(nMS + 255) / 256, 256, 0, stream>>>(Mx, Sx, nMS);
        zero_kernel<<<(int)((SB + 255) / 256), 256, 0, stream>>>(ATT, (int)SB);

        edge_logits_kernel<<<blkEw, 128, 0, stream>>>(Qb_, Kb_, sp_ei, sp_ea, EAM,
                                                      fw_l, fb_l, LG, Mx);
        edge_exp_kernel<<<(nEh + 255) / 256, 256, 0, stream>>>(LG, Mx, Sx, sp_ei);
        edge_scatter_kernel<<<blkEw, 128, 0, stream>>>(Vb_, LG, Sx, sp_ei, ATT);

        // Y = ATT @ ow + ob   (Q buffer reused as Y)
        gemm_wmma_f32<<<gemmGrid, 128, 0, stream>>>(ATT, ow_l, ob + l * H_, Qb_, ROWS, H_, H_);
        // X = LN(X + Y)
        ln_residual_kernel<<<ROWS / 4, 128, 0, stream>>>(X, Qb_, ln_g + l * H_, ln_b + l * H_);
    }

    head_kernel<<<(B_ * N_ + 3) / 4, 128, 0, stream>>>(X, head_w, head_b, out);
}